// SparseMultiHeadAttentionLayer_33217277067652
// MI455X (gfx1250) — compile-verified
//
#include <hip/hip_runtime.h>

// ---------------------------------------------------------------------------
// Banded multi-head attention for MI455X (gfx1250), full-f32 WMMA path.
// D = A(16x4 f32) x B(4x16 f32) + C(16x16 f32)  via v_wmma_f32_16x16x4_f32
// ---------------------------------------------------------------------------

typedef float v2f __attribute__((ext_vector_type(2)));
typedef float v8f __attribute__((ext_vector_type(8)));

#define EMBED   512
#define NHEAD   8
#define DH      64
#define WIN     128
#define BB      4
#define SS      2048
#define NEG_BIG (-1.0e30f)

__device__ __forceinline__ v8f wmma_f32(v2f a, v2f b, v8f c) {
  // 8 args: (neg_a, A, neg_b, B, c_mod, C, reuse_a, reuse_b)
  return __builtin_amdgcn_wmma_f32_16x16x4_f32(false, a, false, b, (short)0, c,
                                               false, false);
}

__device__ __forceinline__ float hmax16(float v) {
  v = fmaxf(v, __shfl_xor(v, 1));
  v = fmaxf(v, __shfl_xor(v, 2));
  v = fmaxf(v, __shfl_xor(v, 4));
  v = fmaxf(v, __shfl_xor(v, 8));
  return v;
}
__device__ __forceinline__ float hsum16(float v) {
  v += __shfl_xor(v, 1);
  v += __shfl_xor(v, 2);
  v += __shfl_xor(v, 4);
  v += __shfl_xor(v, 8);
  return v;
}

// ---------------------------------------------------------------------------
// C[M,N] = A[M,K] @ W[N,K]^T + bias[N]     (one wave per 16x16 C tile)
// A frag (16x4 f32): lane&15 = M row; VGPR0 = K {0|2}, VGPR1 = K {1|3}
// B frag (4x16 f32): lane&15 = N col; VGPR0 = K {0|2}, VGPR1 = K {1|3}
// C frag (8 VGPR):   vgpr r = row r (lanes 0-15) / row 8+r (lanes 16-31)
// ---------------------------------------------------------------------------
__global__ __launch_bounds__(256) void gemm_nt_bias(
    const float* __restrict__ A, const float* __restrict__ W,
    const float* __restrict__ bias, float* __restrict__ C,
    int M, int N, int K) {
  const int lane  = threadIdx.x & 31;
  const int wid   = threadIdx.x >> 5;
  const int l15   = lane & 15;
  const int kb    = (lane >> 4) << 1;   // 0 or 2
  const int rbase = (lane >> 4) << 3;   // 0 or 8

  const int ntiles = N >> 4;
  const int wt = blockIdx.x * 8 + wid;
  const int tm = (wt / ntiles) << 4;
  const int tn = (wt % ntiles) << 4;
  if (tm >= M) return;

  const float* arow = A + (size_t)(tm + l15) * K;
  const float* wrow = W + (size_t)(tn + l15) * K;

  v8f acc = {};
  for (int k0 = 0; k0 < K; k0 += 4) {
    v2f a = *(const v2f*)(arow + k0 + kb);
    v2f b = *(const v2f*)(wrow + k0 + kb);
    acc = wmma_f32(a, b, acc);
  }

  const float bv = bias[tn + l15];
  for (int r = 0; r < 8; ++r)
    C[(size_t)(tm + rbase + r) * N + (tn + l15)] = acc[r] + bv;
}

// ---------------------------------------------------------------------------
// Banded attention.  block = (i-tile, batch), 8 waves = 8 heads.
// qkv layout: [b, s, 3*EMBED]  (q | k | v), head h uses cols h*64..h*64+63
// Writes ctx[b,s,EMBED] and head-averaged weights[b,s,s] (in-band tiles only;
// the rest is pre-zeroed).
// ---------------------------------------------------------------------------
__global__ __launch_bounds__(256) void banded_attn(
    const float* __restrict__ qkv, float* __restrict__ ctx,
    float* __restrict__ wout) {
  __shared__ float lds_attn[NHEAD][16][16];

  const int lane  = threadIdx.x & 31;
  const int h     = threadIdx.x >> 5;
  const int l15   = lane & 15;
  const int kb    = (lane >> 4) << 1;
  const int rbase = (lane >> 4) << 3;
  const int i0    = blockIdx.x << 4;
  const int b     = blockIdx.y;

  const size_t rs = 3 * EMBED;                     // qkv row stride (floats)
  const float* qkvb = qkv + (size_t)b * SS * rs;

  // Q fragments (A-layout), 1/sqrt(dh)=0.125 folded in
  v2f qf[16];
  {
    const float* qrow = qkvb + (size_t)(i0 + l15) * rs + h * DH;
    for (int kk = 0; kk < 16; ++kk) {
      v2f t = *(const v2f*)(qrow + kk * 4 + kb);
      qf[kk].x = t.x * 0.125f;
      qf[kk].y = t.y * 0.125f;
    }
  }

  const int t_lo = max(0, i0 - WIN) >> 4;
  const int t_hi = min(SS - 16, i0 + 15 + WIN) >> 4;

  float mrun[8], lrun[8];
  for (int r = 0; r < 8; ++r) { mrun[r] = NEG_BIG; lrun[r] = 0.0f; }

  // ---- pass 1: online row max / sum over the banded strip -----------------
  for (int t = t_lo; t <= t_hi; ++t) {
    const int j0 = t << 4;
    const float* krow = qkvb + (size_t)(j0 + l15) * rs + EMBED + h * DH;
    v8f s = {};
    for (int kk = 0; kk < 16; ++kk) {
      v2f bf = *(const v2f*)(krow + kk * 4 + kb);
      s = wmma_f32(qf[kk], bf, s);
    }
    const int jg = j0 + l15;
    for (int r = 0; r < 8; ++r) {
      const int ig = i0 + rbase + r;
      int dd = ig - jg; if (dd < 0) dd = -dd;
      float sv = (dd <= WIN) ? s[r] : NEG_BIG;
      const float tmax = hmax16(sv);
      const float mnew = fmaxf(mrun[r], tmax);
      const float ts   = hsum16(__expf(sv - mnew));
      lrun[r] = lrun[r] * __expf(mrun[r] - mnew) + ts;   // self-healing at -1e30
      mrun[r] = mnew;
    }
  }

  float rinv[8];
  for (int r = 0; r < 8; ++r) rinv[r] = 1.0f / lrun[r];

  // ---- pass 2: normalized probs -> weights out + attn @ V -----------------
  v8f acc0 = {}, acc1 = {}, acc2 = {}, acc3 = {};
  for (int t = t_lo; t <= t_hi; ++t) {
    const int j0 = t << 4;
    const float* krow = qkvb + (size_t)(j0 + l15) * rs + EMBED + h * DH;
    v8f s = {};
    for (int kk = 0; kk < 16; ++kk) {
      v2f bf = *(const v2f*)(krow + kk * 4 + kb);
      s = wmma_f32(qf[kk], bf, s);
    }
    const int jg = j0 + l15;
    for (int r = 0; r < 8; ++r) {
      const int ig = i0 + rbase + r;
      int dd = ig - jg; if (dd < 0) dd = -dd;
      const float sv = (dd <= WIN) ? s[r] : NEG_BIG;
      lds_attn[h][rbase + r][l15] = __expf(sv - mrun[r]) * rinv[r];
    }
    __syncthreads();

    // head-averaged attention weights (all 256 threads, one element each)
    {
      const int row = threadIdx.x >> 4;
      const int col = threadIdx.x & 15;
      float sum = 0.0f;
      for (int hh = 0; hh < NHEAD; ++hh) sum += lds_attn[hh][row][col];
      wout[((size_t)b * SS + i0 + row) * SS + (j0 + col)] = sum * 0.125f;
    }

    // ctx += attn_tile(16x16) @ V_tile(16x64): 4 K-chunks x 4 d-chunks
    for (int cc = 0; cc < 4; ++cc) {
      v2f af;
      af.x = lds_attn[h][l15][cc * 4 + kb];
      af.y = lds_attn[h][l15][cc * 4 + kb + 1];
      const float* vbase =
          qkvb + (size_t)(j0 + cc * 4 + kb) * rs + 2 * EMBED + h * DH + l15;
      {
        v2f bf; bf.x = vbase[0];  bf.y = vbase[rs];
        acc0 = wmma_f32(af, bf, acc0);
      }
      {
        v2f bf; bf.x = vbase[16]; bf.y = vbase[rs + 16];
        acc1 = wmma_f32(af, bf, acc1);
      }
      {
        v2f bf; bf.x = vbase[32]; bf.y = vbase[rs + 32];
        acc2 = wmma_f32(af, bf, acc2);
      }
      {
        v2f bf; bf.x = vbase[48]; bf.y = vbase[rs + 48];
        acc3 = wmma_f32(af, bf, acc3);
      }
    }
    __syncthreads();
  }

  // store ctx[b, i, h*64 + d]
  float* crow = ctx + ((size_t)b * SS + i0) * EMBED + h * DH + l15;
  for (int r = 0; r < 8; ++r) {
    const size_t ro = (size_t)(rbase + r) * EMBED;
    crow[ro +  0] = acc0[r];
    crow[ro + 16] = acc1[r];
    crow[ro + 32] = acc2[r];
    crow[ro + 48] = acc3[r];
  }
}

// ---------------------------------------------------------------------------
__global__ __launch_bounds__(256) void zero_f4(float4* __restrict__ p, long n4) {
  const long i = (long)blockIdx.x * 256 + threadIdx.x;
  if (i < n4) p[i] = make_float4(0.f, 0.f, 0.f, 0.f);
}

__global__ __launch_bounds__(256) void maxpool_s(
    const float* __restrict__ outm, float* __restrict__ pooled) {
  const int e = blockIdx.x * 256 + threadIdx.x;   // 0..511
  const int b = blockIdx.y;
  const float* p = outm + (size_t)b * SS * EMBED + e;
  float m = p[0];
  for (int s = 1; s < SS; ++s) m = fmaxf(m, p[(size_t)s * EMBED]);
  pooled[b * EMBED + e] = m;
}

// ---------------------------------------------------------------------------
extern "C" void kernel_launch(void* const* d_in, const int* in_sizes, int n_in,
                              void* d_out, int out_size, void* d_ws,
                              size_t ws_size, hipStream_t stream) {
  const float* emb   = (const float*)d_in[0];  // [4,2048,512]
  const float* w_in  = (const float*)d_in[1];  // [1536,512]
  const float* b_in  = (const float*)d_in[2];  // [1536]
  const float* w_out = (const float*)d_in[3];  // [512,512]
  const float* b_out = (const float*)d_in[4];  // [512]

  float* pooled  = (float*)d_out;              // [4,512]
  float* weights = (float*)d_out + BB * EMBED; // [4,2048,2048]

  float* qkv  = (float*)d_ws;                            // [4,2048,1536]
  float* ctx  = qkv + (size_t)BB * SS * 3 * EMBED;       // [4,2048,512]
  float* outm = ctx + (size_t)BB * SS * EMBED;           // [4,2048,512]

  // 1) zero the dense weights output (out-of-band entries stay 0)
  {
    const long n4 = (long)BB * SS * SS / 4;
    zero_f4<<<(unsigned)((n4 + 255) / 256), 256, 0, stream>>>((float4*)weights, n4);
  }
  // 2) QKV projection: [8192,512] @ [1536,512]^T + b
  {
    const int M = BB * SS, N = 3 * EMBED, K = EMBED;
    const int tiles = (M / 16) * (N / 16);               // 49152
    gemm_nt_bias<<<tiles / 8, 256, 0, stream>>>(emb, w_in, b_in, qkv, M, N, K);
  }
  // 3) banded attention + weights output + ctx
  banded_attn<<<dim3(SS / 16, BB), 256, 0, stream>>>(qkv, ctx, weights);
  // 4) output projection: [8192,512] @ [512,512]^T + b
  {
    const int M = BB * SS, N = EMBED, K = EMBED;
    const int tiles = (M / 16) * (N / 16);               // 16384
    gemm_nt_bias<<<tiles / 8, 256, 0, stream>>>(ctx, w_out, b_out, outm, M, N, K);
  }
  // 5) max-pool over sequence
  maxpool_s<<<dim3(EMBED / 256, BB), 256, 0, stream>>>(outm, pooled);
}